// FloatingRegionScore_31026843746601
// MI455X (gfx1250) — compile-verified
//
#include <hip/hip_runtime.h>
#include <hip/hip_bf16.h>
#include <math.h>

#define CCH 19
#define H 512
#define W 1024
#define HW (H * W)
#define RAD 16
#define KSZ 33
#define EPS 1e-6f
#define INV_LOG_C 0.33962330f  // 1 / ln(19)
#define HT (H / 16)            // 32 row tiles
#define WT (W / 16)            // 64 col tiles
#define NPLANES (1 + CCH)      // entropy + 19 class-count planes

typedef float v2f __attribute__((ext_vector_type(2)));
typedef float v8f __attribute__((ext_vector_type(8)));

// ---------------------------------------------------------------------------
// Kernel 1: per-pixel softmax entropy + argmax label. Memory bound: 40MB read.
// ---------------------------------------------------------------------------
__global__ __launch_bounds__(256) void k_pixel(const float* __restrict__ logit,
                                               float* __restrict__ ent,
                                               int* __restrict__ lbl) {
  int idx = blockIdx.x * blockDim.x + threadIdx.x;
  if (idx >= HW) return;
  float x[CCH];
#pragma unroll
  for (int c = 0; c < CCH; ++c) x[c] = logit[(size_t)c * HW + idx];
  float m = x[0];
  int am = 0;
#pragma unroll
  for (int c = 1; c < CCH; ++c)
    if (x[c] > m) { m = x[c]; am = c; }
  float s = 0.f;
#pragma unroll
  for (int c = 0; c < CCH; ++c) { x[c] = __expf(x[c] - m); s += x[c]; }
  float inv = 1.f / s;
  float e = 0.f;
#pragma unroll
  for (int c = 0; c < CCH; ++c) {
    float p = x[c] * inv;
    e -= p * __logf(p + EPS);
  }
  ent[idx] = e * INV_LOG_C;
  lbl[idx] = am;
}

// ---------------------------------------------------------------------------
// Kernel 2: horizontal 33-wide box pass. Produces rowbuf[20][H][W]:
//   plane 0      = row box-sum of entropy
//   plane 1+c    = per-class count in 33-wide row window (exact f32 ints)
// LDS-tiled so each global element is read once per block.
// ---------------------------------------------------------------------------
__global__ __launch_bounds__(256) void k_row(const float* __restrict__ ent,
                                             const int* __restrict__ lbl,
                                             float* __restrict__ rowbuf) {
  __shared__ float s_e[256 + 2 * RAD];
  __shared__ int s_l[256 + 2 * RAD];
  const int h = blockIdx.y;
  const int w0 = blockIdx.x * 256;
  const int t = threadIdx.x;
  for (int i = t; i < 256 + 2 * RAD; i += 256) {
    int w = w0 - RAD + i;
    bool ok = (w >= 0) && (w < W);
    s_e[i] = ok ? ent[h * W + w] : 0.f;   // zero padding (matches conv pad)
    s_l[i] = ok ? lbl[h * W + w] : -1;    // -1 never matches a class
  }
  __syncthreads();
  const int w = w0 + t;
  float se = 0.f;
#pragma unroll
  for (int d = 0; d < KSZ; ++d) se += s_e[t + d];
  rowbuf[(size_t)h * W + w] = se;
  int ls[KSZ];
#pragma unroll
  for (int d = 0; d < KSZ; ++d) ls[d] = s_l[t + d];
#pragma unroll
  for (int c = 0; c < CCH; ++c) {
    float cnt = 0.f;
#pragma unroll
    for (int d = 0; d < KSZ; ++d) cnt += (ls[d] == c) ? 1.f : 0.f;
    rowbuf[(size_t)(1 + c) * HW + (size_t)h * W + w] = cnt;
  }
}

// ---------------------------------------------------------------------------
// Kernel 3: vertical 33-tall box pass as banded matmul with
// V_WMMA_F32_16X16X4_F32, all 20 planes accumulated per wave, then the
// impurity/uncertainty/score combine done entirely in registers.
//
// Out[16x16 tile t] = sum over source tiles s in {t-1,t,t+1} of  A_s * In_s
// where A_s[h,k] = (|16*t+h - (16*s+k)| <= 16). A is generated on the fly.
// ---------------------------------------------------------------------------
__global__ __launch_bounds__(128) void k_col(const float* __restrict__ rowbuf,
                                             float* __restrict__ out) {
  const int gwave = (int)((blockIdx.x * blockDim.x + threadIdx.x) >> 5);
  const int lane = threadIdx.x & 31;
  const int wt = gwave % WT;
  const int ht = gwave / WT;
  if (ht >= HT) return;  // wave-uniform; grid sized exactly anyway

  v8f acc[NPLANES];
#pragma unroll
  for (int p = 0; p < NPLANES; ++p) acc[p] = (v8f){0.f, 0.f, 0.f, 0.f, 0.f, 0.f, 0.f, 0.f};

  const int half = lane >> 4;       // 0: lanes 0-15, 1: lanes 16-31
  const int mrow = lane & 15;       // A-operand M index / D-operand N index
  const int col = wt * 16 + mrow;   // output/input column
  const int orow = ht * 16 + mrow;  // A-operand output row

  for (int s = ht - 1; s <= ht + 1; ++s) {
    if (s < 0 || s >= HT) continue;  // zero padding at top/bottom
#pragma unroll
    for (int kb = 0; kb < 4; ++kb) {
      // A 16x4 f32 layout: lanes 0-15 hold K={0,1}, lanes 16-31 hold K={2,3}
      const int in0 = s * 16 + kb * 4 + 2 * half;
      v2f a;
      {
        int d0 = in0 - orow;
        int d1 = in0 + 1 - orow;
        a.x = (d0 >= -RAD && d0 <= RAD) ? 1.f : 0.f;
        a.y = (d1 >= -RAD && d1 <= RAD) ? 1.f : 0.f;
      }
      // B 4x16 f32 layout: lanes 0-15 hold K={0,1}, lanes 16-31 hold K={2,3},
      // N = lane & 15.  Same K indexing as A -> same source rows in0, in0+1.
      const size_t boff = (size_t)in0 * W + col;
#pragma unroll
      for (int p = 0; p < NPLANES; ++p) {
        const float* bp = rowbuf + (size_t)p * HW + boff;
        v2f b;
        b.x = bp[0];
        b.y = bp[W];
        acc[p] = __builtin_amdgcn_wmma_f32_16x16x4_f32(
            /*neg_a=*/false, a, /*neg_b=*/false, b,
            /*c_mod=*/(short)0, acc[p], /*reuse_a=*/false, /*reuse_b=*/false);
      }
    }
  }

  // C/D layout: vgpr i holds M = i + 8*half, N = lane&15.  For a fixed
  // (i, lane) all 20 plane accumulators are the SAME pixel -> combine here.
#pragma unroll
  for (int i = 0; i < 8; ++i) {
    float ru = acc[0][i];  // region uncertainty (box sum of entropy)
    float cnt = 0.f;
#pragma unroll
    for (int c = 1; c < NPLANES; ++c) cnt += acc[c][i];
    float invc = 1.f / cnt;  // window always has >= 17x17 valid pixels
    float imp = 0.f;
#pragma unroll
    for (int c = 1; c < NPLANES; ++c) {
      float d = acc[c][i] * invc;
      imp -= d * __logf(d + EPS);
    }
    imp *= INV_LOG_C;
    float pu = ru * invc;
    const int row = ht * 16 + half * 8 + i;
    const size_t o = (size_t)row * W + col;
    out[o] = imp * pu;          // score
    out[HW + o] = imp;          // region_impurity
    out[2 * (size_t)HW + o] = pu;  // prediction_uncertainty
  }
}

// ---------------------------------------------------------------------------
extern "C" void kernel_launch(void* const* d_in, const int* in_sizes, int n_in,
                              void* d_out, int out_size, void* d_ws, size_t ws_size,
                              hipStream_t stream) {
  const float* logit = (const float*)d_in[0];
  float* out = (float*)d_out;

  // Workspace layout: ent [HW f32] | lbl [HW i32] | rowbuf [20*HW f32]
  float* ent = (float*)d_ws;
  int* lbl = (int*)((char*)d_ws + (size_t)HW * sizeof(float));
  float* rowbuf = (float*)((char*)d_ws + 2 * (size_t)HW * sizeof(float));

  k_pixel<<<HW / 256, 256, 0, stream>>>(logit, ent, lbl);

  dim3 rg(W / 256, H, 1);
  k_row<<<rg, 256, 0, stream>>>(ent, lbl, rowbuf);

  // one wave (32 threads) per 16x16 tile: HT*WT = 2048 waves, 4 waves/block
  const int tiles = HT * WT;
  k_col<<<tiles / 4, 128, 0, stream>>>(rowbuf, out);
}